// MaxKGCNConv_11768210391446
// MI455X (gfx1250) — compile-verified
//
#include <hip/hip_runtime.h>
#include <math.h>

typedef __attribute__((ext_vector_type(2))) float v2f;
typedef __attribute__((ext_vector_type(8))) float v8f;

#define IN_FEATS 256
#define OUT_FEATS 64
#define MAXK 16
#define TILE_M 16
#define LDA (IN_FEATS + 4)   // padded LDS stride (260 % 64 banks == 4 -> conflict free)
#define LDH (OUT_FEATS + 4)  // 68

// ---------------- init: d_out = broadcast bias; zero degree arrays ----------
__global__ void init_kernel(float* __restrict__ out, const float* __restrict__ bias,
                            float* __restrict__ deg, int n_nodes) {
    long long i = (long long)blockIdx.x * blockDim.x + threadIdx.x;
    long long total = (long long)n_nodes * OUT_FEATS;
    if (i < total) out[i] = bias[i & (OUT_FEATS - 1)];
    if (i < 2LL * n_nodes) deg[i] = 0.0f;
}

// --------- degrees (float atomics, exact for counts < 2^24) + W transpose ---
__global__ void deg_wt_kernel(const int* __restrict__ src, const int* __restrict__ dst,
                              float* __restrict__ out_deg, float* __restrict__ in_deg,
                              const float* __restrict__ weight, float* __restrict__ wT,
                              int n_edges) {
    int i = blockIdx.x * blockDim.x + threadIdx.x;
    if (i < n_edges) {
        atomicAdd(&out_deg[src[i]], 1.0f);
        atomicAdd(&in_deg[dst[i]], 1.0f);
    }
    if (i < IN_FEATS * OUT_FEATS) {
        int k = i >> 6;          // 0..255
        int c = i & 63;          // 0..63
        wT[c * IN_FEATS + k] = weight[i];
    }
}

// ---------------- fused WMMA GEMM + MaxK(16) + degree scaling ---------------
// 128 threads = 4 waves; block computes rows [m0, m0+16) x all 64 cols.
// Wave w owns the 16x16 C tile for cols [16w, 16w+16), K-loop of 64 x
// V_WMMA_F32_16X16X4_F32. Output compacted: 16 scaled vals + 16 byte indices.
__global__ __launch_bounds__(128)
void gemm_maxk_kernel(const float* __restrict__ feat, const float* __restrict__ wT,
                      const float* __restrict__ out_deg, const float* __restrict__ in_deg,
                      float* __restrict__ tk_vals, unsigned int* __restrict__ tk_idx,
                      int n_nodes) {
    __shared__ float sA[TILE_M * LDA];
    __shared__ float sH[TILE_M * LDH];

    const int tid = threadIdx.x;
    const int m0 = blockIdx.x * TILE_M;

    // cooperative float4 load of the 16x256 feat tile, zero-pad OOB rows
    for (int i = tid; i < TILE_M * (IN_FEATS / 4); i += 128) {
        int row = i / (IN_FEATS / 4);
        int c4  = i % (IN_FEATS / 4);
        float4 v = make_float4(0.f, 0.f, 0.f, 0.f);
        if (m0 + row < n_nodes)
            v = ((const float4*)feat)[(long long)(m0 + row) * (IN_FEATS / 4) + c4];
        float* p = &sA[row * LDA + c4 * 4];
        p[0] = v.x; p[1] = v.y; p[2] = v.z; p[3] = v.w;
    }
    __syncthreads();

    const int wave = tid >> 5;            // 0..3 -> N tile index
    const int lane = tid & 31;
    const int r16  = lane & 15;
    const int koff = (lane >> 4) << 1;    // lanes 0-15: K+0/K+1, lanes 16-31: K+2/K+3
    const float* wcol = wT + (wave * 16 + r16) * IN_FEATS; // column of W (transposed)
    const float* arow = sA + r16 * LDA;

    v8f c = {0.f, 0.f, 0.f, 0.f, 0.f, 0.f, 0.f, 0.f};
    #pragma unroll
    for (int k0 = 0; k0 < IN_FEATS; k0 += 4) {
        v2f a, b;
        a.x = arow[k0 + koff];
        a.y = arow[k0 + koff + 1];
        b.x = wcol[k0 + koff];
        b.y = wcol[k0 + koff + 1];
        // (neg_a, A, neg_b, B, c_mod, C, reuse_a, reuse_b)
        c = __builtin_amdgcn_wmma_f32_16x16x4_f32(false, a, false, b, (short)0, c,
                                                  false, false);
    }

    // spill C tile: VGPR r -> row r (lanes 0-15) / row r+8 (lanes 16-31)
    {
        const int rbase = (lane >> 4) << 3;
        const int col = wave * 16 + r16;
        #pragma unroll
        for (int r = 0; r < 8; ++r)
            sH[(rbase + r) * LDH + col] = c[r];
    }
    __syncthreads();

    // per-row top-16 selection (destructive max-scan in LDS) + scaling
    if (tid < TILE_M && (m0 + tid) < n_nodes) {
        const int node = m0 + tid;
        float od = out_deg[node]; od = od < 1.f ? 1.f : od;
        float id = in_deg[node];  id = id < 1.f ? 1.f : id;
        const float scale = rsqrtf(od) * rsqrtf(id);
        float* rowp = &sH[tid * LDH];

        unsigned int iout0 = 0u, iout1 = 0u, iout2 = 0u, iout3 = 0u;
        #pragma unroll
        for (int i = 0; i < MAXK; ++i) {
            float best = -3.402823466e+38f;
            int bi = 0;
            for (int j = 0; j < OUT_FEATS; ++j) {
                float v = rowp[j];
                if (v > best) { best = v; bi = j; }
            }
            rowp[bi] = -3.402823466e+38f;
            tk_vals[(long long)node * MAXK + i] = best * scale;
            unsigned int b8 = ((unsigned int)bi) << ((i & 3) * 8);
            if (i < 4)       iout0 |= b8;
            else if (i < 8)  iout1 |= b8;
            else if (i < 12) iout2 |= b8;
            else             iout3 |= b8;
        }
        unsigned int* iq = &tk_idx[(long long)node * 4];
        iq[0] = iout0; iq[1] = iout1; iq[2] = iout2; iq[3] = iout3;
    }
}

// ---------------- sparse edge scatter: 4 threads/edge, 4 atomics each -------
__global__ void scatter_kernel(const int* __restrict__ src, const int* __restrict__ dst,
                               const float* __restrict__ tk_vals,
                               const unsigned int* __restrict__ tk_idx,
                               float* __restrict__ out, int n_edges) {
    long long gid = (long long)blockIdx.x * blockDim.x + threadIdx.x;
    long long total = (long long)n_edges * 4;
    if (gid >= total) return;
    int e = (int)(gid >> 2);
    int part = (int)(gid & 3);
    int s = src[e], d = dst[e];
    float4 v = ((const float4*)tk_vals)[(long long)s * 4 + part];
    unsigned int ip = tk_idx[(long long)s * 4 + part];
    float* orow = out + (long long)d * OUT_FEATS;
    atomicAdd(orow + (ip & 0xffu), v.x);
    atomicAdd(orow + ((ip >> 8) & 0xffu), v.y);
    atomicAdd(orow + ((ip >> 16) & 0xffu), v.z);
    atomicAdd(orow + ((ip >> 24) & 0xffu), v.w);
}

extern "C" void kernel_launch(void* const* d_in, const int* in_sizes, int n_in,
                              void* d_out, int out_size, void* d_ws, size_t ws_size,
                              hipStream_t stream) {
    const float* feat   = (const float*)d_in[0];
    const float* weight = (const float*)d_in[1];
    const float* bias   = (const float*)d_in[2];
    const int*   src    = (const int*)d_in[3];
    const int*   dst    = (const int*)d_in[4];
    float* out = (float*)d_out;

    const int n_nodes = in_sizes[0] / IN_FEATS;
    const int n_edges = in_sizes[3];

    // workspace layout (floats): out_deg | in_deg | W^T | topk vals | topk idx
    float* ws_f    = (float*)d_ws;
    float* out_deg = ws_f;
    float* in_deg  = out_deg + n_nodes;
    float* wT      = in_deg + n_nodes;
    float* tk_vals = wT + IN_FEATS * OUT_FEATS;
    unsigned int* tk_idx = (unsigned int*)(tk_vals + (long long)n_nodes * MAXK);

    {
        long long total = (long long)n_nodes * OUT_FEATS;
        int grid = (int)((total + 255) / 256);
        init_kernel<<<grid, 256, 0, stream>>>(out, bias, out_deg, n_nodes);
    }
    {
        int work = n_edges > IN_FEATS * OUT_FEATS ? n_edges : IN_FEATS * OUT_FEATS;
        int grid = (work + 255) / 256;
        deg_wt_kernel<<<grid, 256, 0, stream>>>(src, dst, out_deg, in_deg, weight, wT,
                                                n_edges);
    }
    {
        int grid = (n_nodes + TILE_M - 1) / TILE_M;
        gemm_maxk_kernel<<<grid, 128, 0, stream>>>(feat, wT, out_deg, in_deg,
                                                   tk_vals, tk_idx, n_nodes);
    }
    {
        long long total = (long long)n_edges * 4;
        int grid = (int)((total + 255) / 256);
        scatter_kernel<<<grid, 256, 0, stream>>>(src, dst, tk_vals, tk_idx, out, n_edges);
    }
}